// SubgraphConvolution_37417755082988
// MI455X (gfx1250) — compile-verified
//
#include <hip/hip_runtime.h>

typedef __attribute__((ext_vector_type(2))) float v2f;
typedef __attribute__((ext_vector_type(8))) float v8f;

#define NNODES 100000
#define DDIM   128
#define KNBR   32
#define ROW_LDS 132   // 128 + 4 pad -> A-tile LDS bank = (4*lane + k) % 64, conflict-free

// Pack W^T in K-pair-interleaved layout so each WMMA B fragment is one 8B load:
//   Wp[(k>>1)*256 + n*2 + (k&1)] = W[n][k]  ( = Wt[k][n] )
__global__ void pack_w_kernel(const float* __restrict__ W, float* __restrict__ Wp) {
    int t = blockIdx.x * blockDim.x + threadIdx.x;
    if (t < DDIM * DDIM) {
        int n = t >> 7;        // row of W  (output col)
        int k = t & 127;       // col of W  (K index)
        Wp[(k >> 1) * (2 * DDIM) + n * 2 + (k & 1)] = W[n * DDIM + k];
    }
}

__global__ void __launch_bounds__(128)
subgraph_conv_kernel(const float* __restrict__ feats,
                     const long long* __restrict__ nbr_idx,
                     const long long* __restrict__ node_idx,
                     const float* __restrict__ Wp,
                     const float* __restrict__ bias,
                     float* __restrict__ out) {
    __shared__ float agg_lds[64 * ROW_LDS];   // 33,792 B

    const int lane = threadIdx.x & 31;
    const int wv   = threadIdx.x >> 5;
    const int half = lane >> 4;     // 0: lanes 0-15, 1: lanes 16-31
    const int l16  = lane & 15;
    const int base = blockIdx.x * 64;

    // ---- Phase A: neighbor gather + mean. One wave per node row.
    // All 32 neighbor indices loaded in ONE coalesced b64 load (idx[lane]),
    // then broadcast per step via v_readlane -> SGPR base for the row gather.
    for (int i = 0; i < 16; ++i) {
        const int node = base + wv * 16 + i;
        const int nd   = node < NNODES ? node : NNODES - 1;   // clamp (stores guarded later)
        const int myidx = (int)nbr_idx[(long long)nd * KNBR + lane];
        float a0 = 0.f, a1 = 0.f, a2 = 0.f, a3 = 0.f;
        #pragma unroll 8
        for (int k = 0; k < KNBR; ++k) {
            const int nbr = __builtin_amdgcn_readlane(myidx, k);   // uniform -> SGPR
            const float4 v = *(const float4*)(feats + nbr * DDIM + lane * 4);
            a0 += v.x; a1 += v.y; a2 += v.z; a3 += v.w;
        }
        const float s = 1.0f / (float)KNBR;
        float* dst = &agg_lds[(wv * 16 + i) * ROW_LDS + lane * 4];
        dst[0] = a0 * s; dst[1] = a1 * s; dst[2] = a2 * s; dst[3] = a3 * s;
    }
    __syncthreads();

    // ---- Phase B: D(16x128) = agg(16x128) @ W^T(128x128) via V_WMMA_F32_16X16X4_F32.
    // A 16x4 f32: lanes 0-15 hold M=lane, K=k0+{0,1}; lanes 16-31 M=lane-16, K=k0+{2,3}.
    // B 4x16:     VGPR v holds row K=v (lanes 0-15) / K=v+2 (lanes 16-31), N=lane%16.
    v8f acc[8] = {};   // 8 N-tiles of 16x16 f32 accumulators (64 VGPRs)

    const float* arow = &agg_lds[(wv * 16 + l16) * ROW_LDS];
    #pragma unroll 4
    for (int k0 = 0; k0 < DDIM; k0 += 4) {
        v2f a;
        const float* ap = arow + k0 + 2 * half;
        a.x = ap[0];
        a.y = ap[1];
        // pair index for this lane-half: rows (k0+2*half, k0+2*half+1)
        const v2f* bp = (const v2f*)(Wp + ((k0 >> 1) + half) * (2 * DDIM) + l16 * 2);
        #pragma unroll
        for (int t = 0; t < 8; ++t) {
            const v2f b = bp[t * 16];   // 8B contiguous; 16 lanes -> 128B segment
            acc[t] = __builtin_amdgcn_wmma_f32_16x16x4_f32(
                false, a, false, b, (short)0, acc[t], false, false);
        }
    }

    // ---- Epilogue: out[row] = feats[node_idx[row]] + bias + acc
    // C/D layout: VGPR v -> M = v + 8*half; N = lane%16.
    float bv[8];
    #pragma unroll
    for (int t = 0; t < 8; ++t) bv[t] = bias[t * 16 + l16];

    const int r0 = base + wv * 16;
    #pragma unroll
    for (int v = 0; v < 8; ++v) {
        const int row = r0 + v + 8 * half;
        if (row < NNODES) {
            const int rsrc = (int)node_idx[row];
            const float* fr = feats + rsrc * DDIM;
            float* orow = out + row * DDIM;
            #pragma unroll
            for (int t = 0; t < 8; ++t) {
                const int col = t * 16 + l16;
                orow[col] = fr[col] + bv[t] + acc[t][v];
            }
        }
    }
}

extern "C" void kernel_launch(void* const* d_in, const int* in_sizes, int n_in,
                              void* d_out, int out_size, void* d_ws, size_t ws_size,
                              hipStream_t stream) {
    const float*     feats    = (const float*)d_in[0];
    const long long* nbr_idx  = (const long long*)d_in[1];
    const long long* node_idx = (const long long*)d_in[2];
    const float*     W        = (const float*)d_in[3];
    const float*     b        = (const float*)d_in[4];
    float* out = (float*)d_out;
    float* Wp  = (float*)d_ws;   // 64 KB scratch for pair-packed W^T

    pack_w_kernel<<<(DDIM * DDIM + 255) / 256, 256, 0, stream>>>(W, Wp);

    const int blocks = (NNODES + 63) / 64;   // 64 rows per workgroup (4 waves x 16)
    subgraph_conv_kernel<<<blocks, 128, 0, stream>>>(feats, nbr_idx, node_idx, Wp, b, out);
}